// HistogramLoss_2channel_50044958933593
// MI455X (gfx1250) — compile-verified
//
#include <hip/hip_runtime.h>

// Problem constants (from reference): BIN=256, IM=256, DELTA=0.01
#define IM    256
#define PIX   (IM * IM)        // 65536 pixels per (batch, image)
#define HBIN  128              // only bins [128,255] are reachable for u in [0,1]
#define HSZ   (HBIN * HBIN)    // 16384 entries = 64 KB f32 (fits WGP LDS easily)

typedef __attribute__((ext_vector_type(2))) float v2f;
typedef __attribute__((ext_vector_type(8))) float v8f;

// Triangular soft-bin weight, same arithmetic sequence as the reference:
// relu(1 - |u + (1 - r/128)| * 255); (1 - r/128) is exact in f32 for r<=256.
__device__ __forceinline__ float triw(float u, int r) {
    float bias = 1.0f - (float)r * (1.0f / 128.0f);
    return fmaxf(0.0f, 1.0f - fabsf(u + bias) * 255.0f);
}

// Kernel 1: sparse scatter into an LDS-resident 128x128 f32 histogram.
// blockIdx.x = combo * NP + part, combo = imgSel*2 + batch (4 combos).
// Each pixel contributes <=4 nonzero outer-product entries -> 4 ds_add_f32.
__global__ __launch_bounds__(256)
void hist_scatter(const float* __restrict__ inp, const float* __restrict__ ref,
                  float* __restrict__ ws, int NP, int pixPerWG)
{
    __shared__ float h[HSZ];                       // 64 KB LDS histogram
    for (int i = threadIdx.x; i < HSZ; i += 256) h[i] = 0.0f;
    __syncthreads();

    const int part  = blockIdx.x % NP;
    const int combo = blockIdx.x / NP;
    const int b     = combo & 1;
    const float* img = (combo & 2) ? ref : inp;
    const float* c0  = img + (size_t)b * 2 * PIX;  // channel 0 -> "ha" (r axis)
    const float* c1  = c0 + PIX;                   // channel 1 -> "hb" (j axis)

    const int p0 = part * pixPerWG;
    const int p1 = min(p0 + pixPerWG, PIX);

    for (int p = p0 + (int)threadIdx.x; p < p1; p += 256) {
        float ua = (c0[p] + 1.0f) * 0.5f;
        float ub = (c1[p] + 1.0f) * 0.5f;
        // candidate bins: triangle support radius 1/255 around u = r/128 - 1
        int ra = (int)floorf(ua * 128.0f + 128.0f);
        int rb = (int)floorf(ub * 128.0f + 128.0f);

        float wa0 = triw(ua, ra),     wa1 = triw(ua, ra + 1);
        float wb0 = triw(ub, rb),     wb1 = triw(ub, rb + 1);
        int ia0 = ra - 128, ia1 = ia0 + 1;
        int jb0 = rb - 128, jb1 = jb0 + 1;
        if ((unsigned)ia0 > 127u) { wa0 = 0.0f; ia0 = 0; }
        if ((unsigned)ia1 > 127u) { wa1 = 0.0f; ia1 = 0; }
        if ((unsigned)jb0 > 127u) { wb0 = 0.0f; jb0 = 0; }
        if ((unsigned)jb1 > 127u) { wb1 = 0.0f; jb1 = 0; }

        atomicAdd(&h[jb0 * HBIN + ia0], wb0 * wa0);
        atomicAdd(&h[jb0 * HBIN + ia1], wb0 * wa1);
        atomicAdd(&h[jb1 * HBIN + ia0], wb1 * wa0);
        atomicAdd(&h[jb1 * HBIN + ia1], wb1 * wa1);
    }
    __syncthreads();

    float* dst = ws + (size_t)blockIdx.x * HSZ;    // private partial region
    for (int i = threadIdx.x; i < HSZ; i += 256) dst[i] = h[i];
}

// Kernel 2: partial-histogram reduction via f32 WMMA (ones-vector GEMM)
// + Huber loss. A = ones(16x4) => D[m,n] = sum_k B[k,n] (+C), exact in f32.
// B layout (ISA 7.12.2): v0 holds rows K0|K2 for lane halves, v1 holds K1|K3,
// N = lane&15. So lane l feeds B with partial[k + 2*half + reg] at column n.
// Every lane's D[0] ends up holding the column sum for n = lane&15.
__global__ __launch_bounds__(256)
void hist_huber(const float* __restrict__ ws, float* __restrict__ out, int NP)
{
    const int lane  = threadIdx.x & 31;
    const int wave  = threadIdx.x >> 5;
    const int strip = blockIdx.x * 8 + wave;       // 2048 strips of 16 outputs
    const int b     = strip >> 10;                 // 1024 strips per batch
    const int n     = (strip & 1023) * 16 + (lane & 15);  // inner idx [0,16384)
    const int half  = lane >> 4;

    const float* pin = ws + (size_t)(b)     * NP * HSZ;   // inp combos 0,1
    const float* pgt = ws + (size_t)(2 + b) * NP * HSZ;   // gt  combos 2,3

    float si, sg;
    if ((NP & 3) == 0) {
        v2f ones = {1.0f, 1.0f};
        v8f ai = {0.f, 0.f, 0.f, 0.f, 0.f, 0.f, 0.f, 0.f};
        v8f ag = {0.f, 0.f, 0.f, 0.f, 0.f, 0.f, 0.f, 0.f};
        for (int k = 0; k < NP; k += 4) {
            v2f bi, bg;
            bi.x = pin[(size_t)(k + 2 * half)     * HSZ + n];  // rows K0 / K2
            bi.y = pin[(size_t)(k + 2 * half + 1) * HSZ + n];  // rows K1 / K3
            bg.x = pgt[(size_t)(k + 2 * half)     * HSZ + n];
            bg.y = pgt[(size_t)(k + 2 * half + 1) * HSZ + n];
            ai = __builtin_amdgcn_wmma_f32_16x16x4_f32(
                     false, ones, false, bi, (short)0, ai, false, false);
            ag = __builtin_amdgcn_wmma_f32_16x16x4_f32(
                     false, ones, false, bg, (short)0, ag, false, false);
        }
        si = ai[0]; sg = ag[0];                    // column sum in every lane
    } else {                                       // fallback (tiny ws only)
        si = 0.0f; sg = 0.0f;
        for (int k = 0; k < NP; ++k) {
            si += pin[(size_t)k * HSZ + n];
            sg += pgt[(size_t)k * HSZ + n];
        }
    }

    float hi = si * (1.0f / 65536.0f);             // / (H*W)
    float hg = sg * (1.0f / 65536.0f);
    float mann = fabsf(hi - hg);
    // Huber: 0.5*m^2/DELTA if m<DELTA else m-0.5*DELTA, DELTA=0.01
    float loss = (mann < 0.01f) ? (mann * mann * 50.0f) : (mann - 0.005f);

    if (lane < 16) {                               // lanes 16-31 are duplicates
        int J = n >> 7, R = n & 127;               // j = hb bin, r = ha bin
        out[(size_t)b * (256 * 256) + (size_t)(J + 128) * 256 + (R + 128)] = loss;
    }
}

extern "C" void kernel_launch(void* const* d_in, const int* in_sizes, int n_in,
                              void* d_out, int out_size, void* d_ws, size_t ws_size,
                              hipStream_t stream)
{
    (void)in_sizes; (void)n_in;
    const float* inp = (const float*)d_in[0];
    const float* ref = (const float*)d_in[1];
    float* out = (float*)d_out;
    float* ws  = (float*)d_ws;

    // Choose partials-per-combo from available scratch (deterministic: ws_size
    // is fixed across calls). NP=8 needs 4*8*64KB = 2 MB.
    int NP = 8;
    while (NP > 1 && (size_t)4 * NP * HSZ * sizeof(float) > ws_size) NP >>= 1;
    int pixPerWG = (PIX + NP - 1) / NP;

    // Outer 3 quadrants of the 256x256 output are exactly zero.
    hipMemsetAsync(d_out, 0, (size_t)out_size * sizeof(float), stream);

    hist_scatter<<<4 * NP, 256, 0, stream>>>(inp, ref, ws, NP, pixPerWG);
    hist_huber<<<256, 256, 0, stream>>>(ws, out, NP);
}